// IntegrationGrid_25786983645300
// MI455X (gfx1250) — compile-verified
//
#include <hip/hip_runtime.h>
#include <math.h>

#define N_MOL   8
#define MA      24
#define DESIGN  26
#define RADIAL  15
#define RM_F    5.0f
#define MS      (RADIAL*DESIGN)       /* 390 points per atom  */
#define TOTALP  (MA*MS)               /* 9360 points/molecule */
#define TILES   ((TOTALP + 31)/32)    /* 293 wave tiles       */
#define PI_F    3.14159265358979323846f

typedef float v2f __attribute__((ext_vector_type(2)));
typedef float v8f __attribute__((ext_vector_type(8)));

__global__ __launch_bounds__(32)
void becke_grid_kernel(const float* __restrict__ coords,
                       const float* __restrict__ sphere,
                       const float* __restrict__ sphere_w,
                       float* __restrict__ out_grid,
                       float* __restrict__ out_dv,
                       float* __restrict__ out_w)
{
    __shared__ __align__(16) float s_coords4[32*4];   // atom x,y,z,0 (padded to 32)
    __shared__ float             s_rn2[32];           // |r_j|^2 (padded zeros)
    __shared__ float             s_sph4[DESIGN*4];    // sphere x,y,z,weight
    __shared__ __align__(16) float s_invdm[MA*MA];    // 1/clip(dm,1e-12)
    __shared__ __align__(16) float s_rx[32*MA];       // 32 points x 24 atoms

    const int lane = threadIdx.x;
    const int mol  = blockIdx.x / TILES;
    const int tile = blockIdx.x % TILES;
    const int pb   = tile * 32;

    // ---------------- prologue: molecule constants into LDS ----------------
    if (lane < MA) {
        float cx = coords[(mol*MA + lane)*3 + 0];
        float cy = coords[(mol*MA + lane)*3 + 1];
        float cz = coords[(mol*MA + lane)*3 + 2];
        s_coords4[lane*4+0] = cx; s_coords4[lane*4+1] = cy;
        s_coords4[lane*4+2] = cz; s_coords4[lane*4+3] = 0.f;
        s_rn2[lane] = cx*cx + cy*cy + cz*cz;
    } else {
        s_coords4[lane*4+0] = 0.f; s_coords4[lane*4+1] = 0.f;
        s_coords4[lane*4+2] = 0.f; s_coords4[lane*4+3] = 0.f;
        s_rn2[lane] = 0.f;
    }
    if (lane < DESIGN) {
        s_sph4[lane*4+0] = sphere[lane*3+0];
        s_sph4[lane*4+1] = sphere[lane*3+1];
        s_sph4[lane*4+2] = sphere[lane*3+2];
        s_sph4[lane*4+3] = sphere_w[lane];
    }
    asm volatile("s_wait_dscnt 0x0" ::: "memory");  // single-wave WG: lockstep, just fence LDS

    #pragma unroll 1
    for (int e = lane; e < MA*MA; e += 32) {        // 576/32 = 18 iters, no tail
        int j = e / MA, k = e - j*MA;
        float dx = s_coords4[j*4+0] - s_coords4[k*4+0];
        float dy = s_coords4[j*4+1] - s_coords4[k*4+1];
        float dz = s_coords4[j*4+2] - s_coords4[k*4+2];
        float d  = sqrtf(dx*dx + dy*dy + dz*dz);
        s_invdm[e] = 1.0f / fmaxf(d, 1e-12f);
    }
    asm volatile("s_wait_dscnt 0x0" ::: "memory");

    // ---------------- per-lane grid point ----------------
    const int hi = lane >> 4;        // which 16-lane half
    const int j0 = lane & 15;        // atom column, group 0 (atoms 0..15)
    const int j1 = j0 + 16;          // atom column, group 1 (atoms 16..31, 24+ padded)

    int p     = pb + lane;
    int valid = (p < TOTALP);
    int pc    = valid ? p : (TOTALP - 1);
    int a     = pc / MS;
    int s     = pc - a*MS;
    int ir    = s / DESIGN;
    int isph  = s - ir*DESIGN;

    // Gauss-Chebyshev radial node + Becke transform weight (fp32, as reference)
    float ang  = PI_F * (float)(2*ir + 1) / (2.0f * (float)RADIAL);
    float z    = -cosf(ang);
    float om   = 1.0f - z, op = 1.0f + z;
    float r    = RM_F * op / om;
    float dr   = 2.0f * RM_F / (om*om);
    float w1   = sqrtf(fmaxf(1.0f - z*z, 0.0f)) * dr * (PI_F / (float)RADIAL);
    float wrad = r*r * 4.0f * PI_F * w1;
    float wtv  = wrad * s_sph4[isph*4+3];

    float xx = s_coords4[a*4+0] + r * s_sph4[isph*4+0];
    float xy = s_coords4[a*4+1] + r * s_sph4[isph*4+1];
    float xz = s_coords4[a*4+2] + r * s_sph4[isph*4+2];
    float xn2 = xx*xx + xy*xy + xz*xz;

    // ---------------- WMMA: rx^2 tile = X * (-2 R^T) + (|x|^2 + |r|^2) ----------------
    // B operand (4x16 f32, K=coord component, N=atom): lanes 0-15 hold K=0,1; lanes 16-31 hold K=2,3
    v2f B0, B1;
    {
        float b0a = (hi == 0) ? s_coords4[j0*4+0] : s_coords4[j0*4+2];
        float b0b = (hi == 0) ? s_coords4[j0*4+1] : 0.f;
        B0[0] = -2.0f * b0a;  B0[1] = -2.0f * b0b;
        float b1a = (hi == 0) ? s_coords4[j1*4+0] : s_coords4[j1*4+2];
        float b1b = (hi == 0) ? s_coords4[j1*4+1] : 0.f;
        B1[0] = -2.0f * b1a;  B1[1] = -2.0f * b1b;
    }
    float rn2_0 = s_rn2[j0];
    float rn2_1 = s_rn2[j1];

    // A operand (16x4 f32, M=point, K=component): lanes 0-15 (x,y), lanes 16-31 (z,0)
    float zl  = __shfl(xz, j0, 32);        // z of point (lane&15)         -> for A0 high half
    float xhp = __shfl(xx, j0 + 16, 32);   // x of point 16+(lane&15)      -> for A1 low half
    float yhp = __shfl(xy, j0 + 16, 32);
    v2f A0, A1;
    A0[0] = hi ? zl : xx;   A0[1] = hi ? 0.f : xy;
    A1[0] = hi ? xz : xhp;  A1[1] = hi ? 0.f : yhp;

    // C operand (16x16 f32 layout): c[i] at M = i + 8*hi, N = lane&15
    v8f c00, c01, c10, c11;
    #pragma unroll
    for (int i = 0; i < 8; ++i) {
        float xb0 = __shfl(xn2, i + (hi << 3), 32);        // points pb+0..15
        float xb1 = __shfl(xn2, 16 + i + (hi << 3), 32);   // points pb+16..31
        c00[i] = xb0 + rn2_0;  c01[i] = xb0 + rn2_1;
        c10[i] = xb1 + rn2_0;  c11[i] = xb1 + rn2_1;
    }

    v8f d00 = __builtin_amdgcn_wmma_f32_16x16x4_f32(false, A0, false, B0, (short)0, c00, false, false);
    v8f d01 = __builtin_amdgcn_wmma_f32_16x16x4_f32(false, A0, false, B1, (short)0, c01, false, false);
    v8f d10 = __builtin_amdgcn_wmma_f32_16x16x4_f32(false, A1, false, B0, (short)0, c10, false, false);
    v8f d11 = __builtin_amdgcn_wmma_f32_16x16x4_f32(false, A1, false, B1, (short)0, c11, false, false);

    // stage rx = sqrt(rx^2) into LDS, point-major
    #pragma unroll
    for (int i = 0; i < 8; ++i) {
        int m0 = i + (hi << 3);
        int m1 = m0 + 16;
        s_rx[m0*MA + j0] = sqrtf(fmaxf(d00[i], 0.f));
        s_rx[m1*MA + j0] = sqrtf(fmaxf(d10[i], 0.f));
        if (j0 < 8) {   // atoms 16..23 only
            s_rx[m0*MA + j1] = sqrtf(fmaxf(d01[i], 0.f));
            s_rx[m1*MA + j1] = sqrtf(fmaxf(d11[i], 0.f));
        }
    }
    asm volatile("s_wait_dscnt 0x0" ::: "memory");

    // ---------------- phase 2: one point per lane, Becke cells + outputs ----------------
    if (valid) {
        float rxv[24];
        const float4* rrow = reinterpret_cast<const float4*>(&s_rx[lane*MA]);
        #pragma unroll
        for (int b = 0; b < 6; ++b) {
            float4 q = rrow[b];
            rxv[4*b+0] = q.x; rxv[4*b+1] = q.y; rxv[4*b+2] = q.z; rxv[4*b+3] = q.w;
        }

        float cellsum = 0.f, vnum = 0.f;
        #pragma unroll 1
        for (int j = 0; j < MA; ++j) {
            float rxj  = s_rx[lane*MA + j];
            float cell = 1.f;
            const float4* ivrow = reinterpret_cast<const float4*>(&s_invdm[j*MA]);
            #pragma unroll
            for (int b = 0; b < 6; ++b) {
                float4 iv = ivrow[b];
                float ivc[4] = {iv.x, iv.y, iv.z, iv.w};
                #pragma unroll
                for (int c = 0; c < 4; ++c) {
                    int k = 4*b + c;
                    float mu = (rxj - rxv[k]) * ivc[c];
                    mu = 0.5f*mu*(3.0f - mu*mu);        // softening pass 1
                    mu = 0.5f*mu*(3.0f - mu*mu);        // softening pass 2
                    float muc = 0.5f*(1.0f - mu);
                    cell *= (k == j) ? 1.0f : muc;      // diagonal -> 1
                }
            }
            cellsum += cell;
            if (j == a) vnum = cell;
        }
        float v = vnum / cellsum;

        size_t row = (size_t)mol * TOTALP + (size_t)p;
        out_w[row]        = v * wtv;
        out_grid[row*3+0] = xx;
        out_grid[row*3+1] = xy;
        out_grid[row*3+2] = xz;

        // dv row: 72 contiguous floats, 288B-aligned -> 18 x b128 stores
        float4* dvp = reinterpret_cast<float4*>(out_dv + row * (size_t)(MA*3));
        #pragma unroll
        for (int q = 0; q < 18; ++q) {
            float vals[4];
            #pragma unroll
            for (int cc = 0; cc < 4; ++cc) {
                int e = 4*q + cc;
                int k = e / 3;
                int c = e - 3*k;
                float X = (c == 0) ? xx : ((c == 1) ? xy : xz);
                vals[cc] = X - s_coords4[k*4 + c];
            }
            dvp[q] = make_float4(vals[0], vals[1], vals[2], vals[3]);
        }
    }
}

extern "C" void kernel_launch(void* const* d_in, const int* in_sizes, int n_in,
                              void* d_out, int out_size, void* d_ws, size_t ws_size,
                              hipStream_t stream) {
    (void)in_sizes; (void)n_in; (void)out_size; (void)d_ws; (void)ws_size;
    // inputs: [0]=labels (unused), [1]=coords, [2]=sphere, [3]=sphere_weights
    const float* coords   = (const float*)d_in[1];
    const float* sphere   = (const float*)d_in[2];
    const float* sphere_w = (const float*)d_in[3];

    float* out = (float*)d_out;
    const size_t grid_elems = (size_t)N_MOL * TOTALP * 3;
    const size_t dv_elems   = (size_t)N_MOL * TOTALP * MA * 3;
    float* out_grid = out;
    float* out_dv   = out + grid_elems;
    float* out_w    = out + grid_elems + dv_elems;

    dim3 grid(N_MOL * TILES);   // 8 * 293 = 2344 single-wave workgroups
    dim3 block(32);
    hipLaunchKernelGGL(becke_grid_kernel, grid, block, 0, stream,
                       coords, sphere, sphere_w, out_grid, out_dv, out_w);
}